// ViterbiLoss_41523743818140
// MI455X (gfx1250) — compile-verified
//
#include <hip/hip_runtime.h>
#include <stdint.h>

// Problem constants (from reference)
#define B_SZ      128
#define T_SZ      512
#define K_SZ      24
#define KK        576          // K*K
#define TILE_BYTES 2304        // KK * 4
#define START_TAG 22
#define STOP_TAG  23

// Async prefetch pipeline
#define STAGES        6
#define OPS_PER_TILE  5        // 4x b128 (512B each) + 1x b64 (256B) = 2304B
#define WAIT_STEADY   ((STAGES - 1) * OPS_PER_TILE)   // 25 outstanding ops => oldest tile done

// ---------------------------------------------------------------------------
// gfx1250 async global->LDS copy of one contiguous 2304-byte tile (wave32).
// the LDS address comes from the VDST VGPR, so we pass the workgroup-relative
// LDS byte offset (low 32 bits of the generic shared pointer).
// Tracked by ASYNCcnt; completion enforced with s_wait_asynccnt.
// ---------------------------------------------------------------------------
__device__ __forceinline__ void async_tile_load(uint32_t lds_off,
                                                const float* __restrict__ gsrc,
                                                int lane) {
  uint64_t g = (uint64_t)(uintptr_t)gsrc;
#pragma unroll
  for (int r = 0; r < 4; ++r) {
    uint32_t l = lds_off + r * 512 + lane * 16;
    uint64_t a = g + r * 512 + lane * 16;
    asm volatile("global_load_async_to_lds_b128 %0, %1, off"
                 :: "v"(l), "v"(a) : "memory");
  }
  {
    uint32_t l = lds_off + 2048 + lane * 8;
    uint64_t a = g + 2048 + lane * 8;
    asm volatile("global_load_async_to_lds_b64 %0, %1, off"
                 :: "v"(l), "v"(a) : "memory");
  }
}

__device__ __forceinline__ void wait_async_steady() {
  asm volatile("s_wait_asynccnt 25" ::: "memory");
}
__device__ __forceinline__ void wait_async_all() {
  asm volatile("s_wait_asynccnt 0" ::: "memory");
}

// ---------------------------------------------------------------------------
// Forward recursion: one wave32 workgroup per batch element.
// s_t[j] = m + log( sum_i exp(s_{t-1}[i] - m) * exp(f[t][i][j]) ),  m = max_i s
// Masked timesteps (t >= length) are identity, so the loop just stops early.
// ---------------------------------------------------------------------------
__global__ __launch_bounds__(32) void crf_forward_kernel(
    const float* __restrict__ features,   // [B, T, K, K]
    const int*   __restrict__ lengths,    // [B]
    float*       __restrict__ finals) {   // [B] -> s[STOP_TAG]
  __shared__ float tiles[STAGES * KK];
  __shared__ float pArr[32];

  const int b    = blockIdx.x;
  const int lane = threadIdx.x;
  const int len  = lengths[b];
  const float* fb = features + (size_t)b * T_SZ * KK;

  const float NEG_INF = -3.0e38f;
  // init = features[b, 0, START_TAG, :]
  float s = (lane < K_SZ) ? fb[START_TAG * K_SZ + lane] : NEG_INF;

  const uint32_t lds_base = (uint32_t)(uintptr_t)(&tiles[0]);
  const int nt = len - 1;                        // tiles to process: t = 1..len-1
  const int jj = (lane < K_SZ) ? lane : 0;       // lanes 24..31 compute a dummy column

  // Prologue: issue tiles 1..min(STAGES-1, nt); tile t lives in buf (t-1)%STAGES
  const int pro = (nt < (STAGES - 1)) ? nt : (STAGES - 1);
  for (int k = 0; k < pro; ++k)
    async_tile_load(lds_base + (uint32_t)(k % STAGES) * TILE_BYTES,
                    fb + (size_t)(k + 1) * KK, lane);

  for (int t = 1; t <= nt; ++t) {
    // Issue tile t+STAGES-1 into the buffer consumed at iteration t-1 (free now).
    const int nxt = t + STAGES - 1;
    if (nxt <= nt)
      async_tile_load(lds_base + (uint32_t)((nxt - 1) % STAGES) * TILE_BYTES,
                      fb + (size_t)nxt * KK, lane);

    // Guarantee tile t has landed in LDS (async ops complete in order).
    if (nt - t >= STAGES - 1) wait_async_steady();
    else                      wait_async_all();

    const float* __restrict__ tile = &tiles[((t - 1) % STAGES) * KK];

    // m = max over lanes of s (lanes 24..31 hold -inf)
    float m = s;
#pragma unroll
    for (int o = 16; o > 0; o >>= 1) {
      float v = __shfl_xor(m, o, 32);
      m = (v > m) ? v : m;
    }
    const float p = __expf(s - m);     // lanes >= 24: exp(-inf) = 0
    pArr[lane] = p;
    __syncthreads();                   // single-wave WG: barrier is a NOP + dscnt wait

    float acc = 0.0f;
#pragma unroll 6
    for (int i = 0; i < K_SZ; ++i)
      acc += pArr[i] * __expf(tile[i * K_SZ + jj]);
    __syncthreads();

    const float ns = m + __logf(acc);  // acc >= exp(f_row_of_max) > 0
    if (lane < K_SZ) s = ns;
  }

  if (lane == STOP_TAG) finals[b] = s;
}

// ---------------------------------------------------------------------------
// Gold score: sum over valid (b,t) of features[b,t].flat[targets[b,t]].
// ---------------------------------------------------------------------------
__global__ void init_ws_kernel(float* __restrict__ w) {
  if (threadIdx.x == 0) w[0] = 0.0f;
}

__global__ __launch_bounds__(256) void gold_kernel(
    const float* __restrict__ features,
    const int*   __restrict__ targets,
    const int*   __restrict__ lengths,
    float*       __restrict__ w) {
  const int b   = blockIdx.x;
  const int len = lengths[b];
  const float* fb = features + (size_t)b * T_SZ * KK;
  const int*   tb = targets  + (size_t)b * T_SZ;

  float acc = 0.0f;
  for (int t = threadIdx.x; t < len; t += blockDim.x)
    acc += fb[(size_t)t * KK + tb[t]];

  __shared__ float red[256];
  red[threadIdx.x] = acc;
  __syncthreads();
#pragma unroll
  for (int o = 128; o > 0; o >>= 1) {
    if (threadIdx.x < o) red[threadIdx.x] += red[threadIdx.x + o];
    __syncthreads();
  }
  if (threadIdx.x == 0) atomicAdd(&w[0], red[0]);
}

// ---------------------------------------------------------------------------
// out[0] = sum_b finals[b] - gold
// ---------------------------------------------------------------------------
__global__ __launch_bounds__(128) void finalize_kernel(
    const float* __restrict__ w, float* __restrict__ out) {
  __shared__ float red[128];
  red[threadIdx.x] = w[1 + threadIdx.x];
  __syncthreads();
#pragma unroll
  for (int o = 64; o > 0; o >>= 1) {
    if (threadIdx.x < o) red[threadIdx.x] += red[threadIdx.x + o];
    __syncthreads();
  }
  if (threadIdx.x == 0) out[0] = red[0] - w[0];
}

// ---------------------------------------------------------------------------
extern "C" void kernel_launch(void* const* d_in, const int* in_sizes, int n_in,
                              void* d_out, int out_size, void* d_ws, size_t ws_size,
                              hipStream_t stream) {
  const float* features = (const float*)d_in[0];   // [B,T,K,K] f32
  const int*   targets  = (const int*)d_in[1];     // [B,T] i32 (jax default x64 off)
  const int*   lengths  = (const int*)d_in[2];     // [B] i32
  float* w   = (float*)d_ws;    // w[0] = gold accum, w[1..B] = per-batch finals
  float* out = (float*)d_out;

  init_ws_kernel <<<1,    32,  0, stream>>>(w);
  gold_kernel    <<<B_SZ, 256, 0, stream>>>(features, targets, lengths, w);
  crf_forward_kernel<<<B_SZ, 32, 0, stream>>>(features, lengths, w + 1);
  finalize_kernel<<<1,   128,  0, stream>>>(w, out);
}